// DrugRepurposingGNN_14937896255532
// MI455X (gfx1250) — compile-verified
//
#include <hip/hip_runtime.h>
#include <hip/hip_bf16.h>

// GraphSAGE 4-layer forward on gfx1250 (MI455X).
//  - Edge aggregation: wave-per-edge coalesced gather + f32 atomic scatter (L2-resident).
//  - Projections: V_WMMA_F32_16X16X4_F32, wave computes a 16x128 output stripe,
//    fused  out = x@Wr + (agg*invdeg)@Wl + b  (+ReLU), fp32 exact.

#define N_NODES 100000
#define N_EDGES 1600000
#define F_INPUT 16
#define HDIM 128
#define M_TILES (N_NODES / 16) // 6250, exact

typedef float v2f __attribute__((ext_vector_type(2)));
typedef float v8f __attribute__((ext_vector_type(8)));

__global__ __launch_bounds__(256) void zero_f4_kernel(float4* __restrict__ p, long long n4) {
    long long i = (long long)blockIdx.x * 256 + threadIdx.x;
    if (i < n4) p[i] = make_float4(0.f, 0.f, 0.f, 0.f);
}

__global__ __launch_bounds__(256) void degree_kernel(const int* __restrict__ dst,
                                                     float* __restrict__ cnt, int E) {
    int e = blockIdx.x * 256 + threadIdx.x;
    if (e < E) atomicAdd(&cnt[dst[e]], 1.0f);
}

__global__ __launch_bounds__(256) void recip_kernel(float* __restrict__ deg, int n) {
    int i = blockIdx.x * 256 + threadIdx.x;
    if (i < n) deg[i] = 1.0f / fmaxf(deg[i], 1.0f);
}

// agg[dst] += in[src]; CHUNKS = K/4 float4 chunks per edge.
// Consecutive lanes handle consecutive chunks of the SAME edge -> coalesced
// 512B gather rows (K=128) and conflict-free consecutive atomic targets.
template <int CHUNKS>
__global__ __launch_bounds__(256) void scatter_kernel(const float* __restrict__ in,
                                                      const int* __restrict__ src,
                                                      const int* __restrict__ dst,
                                                      float* __restrict__ agg, int E) {
    long long tid = (long long)blockIdx.x * 256 + threadIdx.x;
    if (tid >= (long long)E * CHUNKS) return;
    int e = (int)(tid / CHUNKS);
    int c = (int)(tid % CHUNKS);
    int s = src[e];
    int d = dst[e];
    float4 v = ((const float4*)in)[(long long)s * CHUNKS + c];
    float* p = agg + ((long long)d * CHUNKS + c) * 4;
    atomicAdd(p + 0, v.x);
    atomicAdd(p + 1, v.y);
    atomicAdd(p + 2, v.z);
    atomicAdd(p + 3, v.w);
}

// out[16x128 stripe per wave] = in@Wr + (agg*invdeg)@Wl + bias (+relu)
// A frag (16x4 f32): lane m=lane&15 holds row row0+m, K offset kh=(lane>=16)*2,
//   contiguous float2 {K=k+kh, K=k+kh+1}.
// B frag (4x16 f32): lane supplies col = nt*16+m, rows K=k+kh and k+kh+1.
// C/D (16x16 f32, 8 VGPRs): VGPR j -> M = j + (lane>=16)*8, N = lane&15.
__global__ __launch_bounds__(256) void sage_gemm_kernel(const float* __restrict__ in,
                                                        const float* __restrict__ agg,
                                                        const float* __restrict__ invdeg,
                                                        const float* __restrict__ Wr,
                                                        const float* __restrict__ Wl,
                                                        const float* __restrict__ bias,
                                                        float* __restrict__ out,
                                                        int K, int do_relu) {
    const int lane = threadIdx.x & 31;
    const int wv = threadIdx.x >> 5;
    const int mt = blockIdx.x * 8 + wv;
    if (mt >= M_TILES) return; // wave-uniform: EXEC stays all-1s for WMMA

    const int row0 = mt << 4;
    const int m = lane & 15;
    const int kh = (lane >> 4) << 1; // 0 or 2
    const long long arow = (long long)(row0 + m);

    v8f acc[8] = {};

    // pass 1: in @ Wr
    for (int k = 0; k < K; k += 4) {
        v2f a = *(const v2f*)(in + arow * K + k + kh);
#pragma unroll
        for (int nt = 0; nt < 8; ++nt) {
            const int col = (nt << 4) + m;
            v2f b;
            b.x = Wr[(k + kh) * HDIM + col];
            b.y = Wr[(k + kh + 1) * HDIM + col];
            acc[nt] = __builtin_amdgcn_wmma_f32_16x16x4_f32(
                false, a, false, b, (short)0, acc[nt], false, false);
        }
    }

    // pass 2: (agg * invdeg) @ Wl   (mean aggregation folded into A fragment)
    const float inv = invdeg[row0 + m];
    for (int k = 0; k < K; k += 4) {
        v2f a = *(const v2f*)(agg + arow * K + k + kh);
        a.x *= inv;
        a.y *= inv;
#pragma unroll
        for (int nt = 0; nt < 8; ++nt) {
            const int col = (nt << 4) + m;
            v2f b;
            b.x = Wl[(k + kh) * HDIM + col];
            b.y = Wl[(k + kh + 1) * HDIM + col];
            acc[nt] = __builtin_amdgcn_wmma_f32_16x16x4_f32(
                false, a, false, b, (short)0, acc[nt], false, false);
        }
    }

    // epilogue: bias + relu + store (C/D layout)
    const int rofs = (lane >> 4) << 3; // 0 or 8
#pragma unroll
    for (int nt = 0; nt < 8; ++nt) {
        const int col = (nt << 4) + m;
        const float bv = bias[col];
#pragma unroll
        for (int j = 0; j < 8; ++j) {
            float v = acc[nt][j] + bv;
            if (do_relu) v = fmaxf(v, 0.0f);
            out[(long long)(row0 + rofs + j) * HDIM + col] = v;
        }
    }
}

extern "C" void kernel_launch(void* const* d_in, const int* in_sizes, int n_in,
                              void* d_out, int out_size, void* d_ws, size_t ws_size,
                              hipStream_t stream) {
    (void)in_sizes; (void)n_in; (void)out_size; (void)ws_size;

    const float* x = (const float*)d_in[0];
    const int* ei = (const int*)d_in[1]; // [2, E]: row 0 = src, row 1 = dst
    const int* src = ei;
    const int* dst = ei + N_EDGES;
    const float* Wl[4] = {(const float*)d_in[2], (const float*)d_in[5],
                          (const float*)d_in[8], (const float*)d_in[11]};
    const float* Wr[4] = {(const float*)d_in[3], (const float*)d_in[6],
                          (const float*)d_in[9], (const float*)d_in[12]};
    const float* bs[4] = {(const float*)d_in[4], (const float*)d_in[7],
                          (const float*)d_in[10], (const float*)d_in[13]};

    // workspace: agg[N*H] | hA[N*H] | hB[N*H] | deg[N]   (~154 MB)
    float* agg = (float*)d_ws;
    float* hA = agg + (size_t)N_NODES * HDIM;
    float* hB = hA + (size_t)N_NODES * HDIM;
    float* deg = hB + (size_t)N_NODES * HDIM;

    // in-degree -> 1/max(deg,1), computed once (dst identical across layers)
    zero_f4_kernel<<<(N_NODES / 4 + 255) / 256, 256, 0, stream>>>((float4*)deg, N_NODES / 4);
    degree_kernel<<<(N_EDGES + 255) / 256, 256, 0, stream>>>(dst, deg, N_EDGES);
    recip_kernel<<<(N_NODES + 255) / 256, 256, 0, stream>>>(deg, N_NODES);

    const int gemm_grid = (M_TILES + 7) / 8; // 8 waves/block, 1 stripe/wave

    // ---- layer 1: K = 16, relu ----
    {
        long long n4 = (long long)N_NODES * F_INPUT / 4;
        zero_f4_kernel<<<(int)((n4 + 255) / 256), 256, 0, stream>>>((float4*)agg, n4);
        long long t = (long long)N_EDGES * (F_INPUT / 4);
        scatter_kernel<F_INPUT / 4><<<(int)((t + 255) / 256), 256, 0, stream>>>(
            x, src, dst, agg, N_EDGES);
        sage_gemm_kernel<<<gemm_grid, 256, 0, stream>>>(x, agg, deg, Wr[0], Wl[0], bs[0],
                                                        hA, F_INPUT, 1);
    }

    // ---- layers 2..4: K = 128; relu after 2 and 3, not after 4 ----
    const float* cur = hA;
    float* nxt = hB;
    for (int l = 1; l < 4; ++l) {
        long long n4 = (long long)N_NODES * HDIM / 4;
        zero_f4_kernel<<<(int)((n4 + 255) / 256), 256, 0, stream>>>((float4*)agg, n4);
        long long t = (long long)N_EDGES * (HDIM / 4);
        scatter_kernel<HDIM / 4><<<(int)((t + 255) / 256), 256, 0, stream>>>(
            cur, src, dst, agg, N_EDGES);
        float* outp = (l == 3) ? (float*)d_out : nxt;
        int relu = (l == 3) ? 0 : 1;
        sage_gemm_kernel<<<gemm_grid, 256, 0, stream>>>(cur, agg, deg, Wr[l], Wl[l], bs[l],
                                                        outp, HDIM, relu);
        float* t2 = (float*)cur;
        cur = outp;
        nxt = t2;
    }
}